// LuongAttention_47416438948457
// MI455X (gfx1250) — compile-verified
//
#include <hip/hip_runtime.h>

typedef __attribute__((ext_vector_type(2))) float v2f;
typedef __attribute__((ext_vector_type(8))) float v8f;

#define B_ 16
#define S_ 4096
#define D_ 1024
#define CHUNKS 32
#define ROWS 128            // S rows per partial-attention workgroup
#define TILE 16             // rows per LDS tile (64KB), double-buffered
#define NT (ROWS / TILE)    // 8 tiles per workgroup

// ---- CDNA5 async global->LDS copy (ASYNCcnt-tracked, bypasses VGPRs) ------
__device__ __forceinline__ void async_load_b128(unsigned lds_off, const float* g) {
    // VDST = LDS byte address VGPR, VADDR = 64-bit global address, saddr off
    asm volatile("global_load_async_to_lds_b128 %0, %1, off"
                 :: "v"(lds_off), "v"(g) : "memory");
}
template <int N>
__device__ __forceinline__ void wait_asynccnt() {
    asm volatile("s_wait_asynccnt %0" :: "n"(N) : "memory");
}

// ---------------------------------------------------------------------------
// Kernel 1: q[16,1024] = contexts[16,1024] @ W[1024,1024]
// M=16 == batch dim -> V_WMMA_F32_16X16X4_F32, one wave per 16-col tile.
// ---------------------------------------------------------------------------
__global__ void __launch_bounds__(32)
qproj_wmma(const float* __restrict__ contexts,
           const float* __restrict__ W,
           float* __restrict__ q) {
    const int lane  = threadIdx.x;
    const int half  = lane >> 4;
    const int mn    = lane & 15;
    const int nbase = blockIdx.x * 16;

    v8f c = {};
    for (int k0 = 0; k0 < D_; k0 += 4) {
        const int k = k0 + 2 * half;           // lanes 16-31 hold K+2
        v2f a, bm;
        a.x  = contexts[mn * D_ + k];
        a.y  = contexts[mn * D_ + k + 1];
        bm.x = W[(size_t)k       * D_ + nbase + mn];
        bm.y = W[(size_t)(k + 1) * D_ + nbase + mn];
        c = __builtin_amdgcn_wmma_f32_16x16x4_f32(
                false, a, false, bm, (short)0, c, false, false);
    }
    for (int r = 0; r < 8; ++r)
        q[(r + 8 * half) * D_ + nbase + mn] = c[r];
}

// ---------------------------------------------------------------------------
// Kernel 2: single-pass online-softmax partial attention.
// 512 blocks (16 batches x 32 chunks) x 256 threads.  values tiles are
// double-buffered through LDS via global_load_async_to_lds_b128; each HBM
// byte is read exactly once, consumed twice from LDS (score + accumulate).
// ---------------------------------------------------------------------------
__global__ void __launch_bounds__(256)
attn_partial(const float* __restrict__ values,
             const float* __restrict__ q,
             float* __restrict__ pm,
             float* __restrict__ pl,
             float* __restrict__ pacc) {
    __shared__ __attribute__((aligned(16))) float vbuf[2 * TILE * D_]; // 128KB
    __shared__ float sc[TILE];

    const int tid   = threadIdx.x;
    const int wave  = tid >> 5;
    const int lane  = tid & 31;
    const int b     = blockIdx.x >> 5;          // / CHUNKS
    const int chunk = blockIdx.x & (CHUNKS - 1);
    const int s0    = chunk * ROWS;
    const float* vb = values + (size_t)b * S_ * D_;
    const unsigned vbase = (unsigned)(size_t)vbuf;   // LDS byte offset

    // ---- q[b,:] into registers: lane only ever needs elems (i*32+lane)*4 ----
    float4 qv[8];
    {
        const float4* q4 = (const float4*)(q + b * D_);
        #pragma unroll
        for (int i = 0; i < 8; ++i) qv[i] = q4[i * 32 + lane];
    }

    // ---- async-copy one 16-row tile; wave w owns rows w and w+8 ------------
    auto issue_tile = [&](int t, int buf) {
        #pragma unroll
        for (int rr = 0; rr < 2; ++rr) {
            const int r = wave + 8 * rr;
            const float* gsrc = vb + (size_t)(s0 + t * TILE + r) * D_;
            const unsigned lbase =
                vbase + ((unsigned)(buf * TILE + r) * D_) * 4u;
            #pragma unroll
            for (int i = 0; i < 8; ++i) {
                const unsigned e = (unsigned)(i * 32 + lane) * 4u;
                async_load_b128(lbase + e * 4u, gsrc + e);
            }
        }
    };

    float m = -3.402823466e38f;
    float l = 0.f;
    float ax = 0.f, ay = 0.f, az = 0.f, aw = 0.f;
    const int d0 = tid * 4;
    const float4* vb4 = (const float4*)vbuf;

    issue_tile(0, 0);
    for (int t = 0; t < NT; ++t) {
        const int cur = t & 1;
        if (t + 1 < NT) {
            issue_tile(t + 1, cur ^ 1);
            wait_asynccnt<16>();     // tile t's 16 per-wave copies complete
        } else {
            wait_asynccnt<0>();
        }
        __syncthreads();             // tile t visible block-wide

        // ---- scores for tile t (wave w -> rows w, w+8) ----
        #pragma unroll
        for (int rr = 0; rr < 2; ++rr) {
            const int r = wave + 8 * rr;
            float sum = 0.f;
            #pragma unroll
            for (int i = 0; i < 8; ++i) {
                const float4 v = vb4[(cur * TILE + r) * (D_ / 4) + i * 32 + lane];
                sum += v.x * qv[i].x + v.y * qv[i].y
                     + v.z * qv[i].z + v.w * qv[i].w;
            }
            #pragma unroll
            for (int off = 16; off > 0; off >>= 1)
                sum += __shfl_xor(sum, off, 32);
            if (lane == 0) sc[r] = sum;
        }
        __syncthreads();

        // ---- online softmax update (identical on every thread) ----
        float tmax = sc[0];
        #pragma unroll
        for (int r = 1; r < TILE; ++r) tmax = fmaxf(tmax, sc[r]);
        const float newm  = fmaxf(m, tmax);
        const float scale = __expf(m - newm);
        l *= scale; ax *= scale; ay *= scale; az *= scale; aw *= scale;
        #pragma unroll
        for (int r = 0; r < TILE; ++r) {
            const float w  = __expf(sc[r] - newm);
            const float4 v = vb4[(cur * TILE + r) * (D_ / 4) + tid];
            l  += w;
            ax += w * v.x; ay += w * v.y; az += w * v.z; aw += w * v.w;
        }
        m = newm;
        __syncthreads();             // safe to overwrite buffer `cur` at t+2
    }

    float4 o; o.x = ax; o.y = ay; o.z = az; o.w = aw;
    ((float4*)(pacc + (size_t)blockIdx.x * D_))[tid] = o;
    if (tid == 0) { pm[blockIdx.x] = m; pl[blockIdx.x] = l; }
    (void)d0;
}

// ---------------------------------------------------------------------------
// Kernel 3: combine 32 partials per batch (flash-decoding reduction).
// ---------------------------------------------------------------------------
__global__ void __launch_bounds__(256)
attn_reduce(const float* __restrict__ pm,
            const float* __restrict__ pl,
            const float* __restrict__ pacc,
            float* __restrict__ out) {
    const int b   = blockIdx.x;
    const int tid = threadIdx.x;

    float M = -3.402823466e38f;
    for (int c = 0; c < CHUNKS; ++c) M = fmaxf(M, pm[b * CHUNKS + c]);
    float L = 0.f;
    for (int c = 0; c < CHUNKS; ++c)
        L += __expf(pm[b * CHUNKS + c] - M) * pl[b * CHUNKS + c];

    const int d0 = tid * 4;
    float ax = 0.f, ay = 0.f, az = 0.f, aw = 0.f;
    for (int c = 0; c < CHUNKS; ++c) {
        const float s  = __expf(pm[b * CHUNKS + c] - M);
        const float4 v = *(const float4*)(pacc + (size_t)(b * CHUNKS + c) * D_ + d0);
        ax += s * v.x; ay += s * v.y; az += s * v.z; aw += s * v.w;
    }
    const float inv = 1.f / L;
    float4 o; o.x = ax * inv; o.y = ay * inv; o.z = az * inv; o.w = aw * inv;
    *(float4*)(out + b * D_ + d0) = o;
}

// ---------------------------------------------------------------------------
extern "C" void kernel_launch(void* const* d_in, const int* in_sizes, int n_in,
                              void* d_out, int out_size, void* d_ws, size_t ws_size,
                              hipStream_t stream) {
    const float* values   = (const float*)d_in[0];   // [16,4096,1024]
    const float* contexts = (const float*)d_in[1];   // [16,1024]
    const float* W        = (const float*)d_in[2];   // [1024,1024]
    // d_in[3] (bias) cancels under softmax shift-invariance.

    float* ws   = (float*)d_ws;
    float* q    = ws;                         // 16*1024 floats
    float* pm   = q  + B_ * D_;               // 512 floats
    float* pl   = pm + B_ * CHUNKS;           // 512 floats
    float* pacc = pl + B_ * CHUNKS;           // 512*1024 floats (~2MB)

    hipLaunchKernelGGL(qproj_wmma,   dim3(D_ / 16),     dim3(32),  0, stream,
                       contexts, W, q);
    hipLaunchKernelGGL(attn_partial, dim3(B_ * CHUNKS), dim3(256), 0, stream,
                       values, q, pm, pl, pacc);
    hipLaunchKernelGGL(attn_reduce,  dim3(B_),          dim3(256), 0, stream,
                       pm, pl, pacc, (float*)d_out);
}